// SAGESimCorConv_88888643158467
// MI455X (gfx1250) — compile-verified
//
#include <hip/hip_runtime.h>

#define D_FEAT 128

// ---------------------------------------------------------------------------
// Kernel: one wave32 per edge (FIRST in file so its codegen shows in the
// disasm snippet: expect global_atomic_add_f32, global_prefetch_b8, NT loads).
//
// Lanes cooperatively read the 128-float source row as 32 x float4 (512B fully
// coalesced) and scatter-accumulate into nsum[dst] with native f32 L2 atomics.
// Lane 0 bumps the degree counter. Edge-list loads are non-temporal (streamed;
// keep them from evicting the L2-resident feature table). A gfx1250
// global_prefetch pulls a future edge's random source row toward L2.
// ---------------------------------------------------------------------------
__global__ __launch_bounds__(256) void sage_edge_scatter_kernel(
    const float* __restrict__ feat,
    const int* __restrict__ src,
    const int* __restrict__ dst,
    float* __restrict__ nsum,
    float* __restrict__ deg,
    int n_edges) {
    const int lane = threadIdx.x & 31;
    const int e = blockIdx.x * (blockDim.x >> 5) + (threadIdx.x >> 5);
    if (e >= n_edges) return;

    const int s = __builtin_nontemporal_load(src + e);
    const int d = __builtin_nontemporal_load(dst + e);

    // Prefetch a future edge's source row (random access) toward L2.
    const int ep = e + 1024;
    if (ep < n_edges) {
        const int sp = __builtin_nontemporal_load(src + ep);
        __builtin_prefetch(feat + (size_t)sp * D_FEAT, 0, 1);
    }

    const float4 v =
        *reinterpret_cast<const float4*>(feat + (size_t)s * D_FEAT + lane * 4);
    float* op = nsum + (size_t)d * D_FEAT + lane * 4;

    // Non-returning f32 atomic adds -> global_atomic_add_f32 (STOREcnt path).
    unsafeAtomicAdd(op + 0, v.x);
    unsafeAtomicAdd(op + 1, v.y);
    unsafeAtomicAdd(op + 2, v.z);
    unsafeAtomicAdd(op + 3, v.w);

    if (lane == 0) {
        unsafeAtomicAdd(deg + d, 1.0f);
    }
}

// ---------------------------------------------------------------------------
// Kernel: zero the accumulator (d_out, reused as nsum) and the degree array.
// ---------------------------------------------------------------------------
__global__ __launch_bounds__(256) void sage_init_kernel(float4* __restrict__ out4,
                                                        float* __restrict__ deg,
                                                        int n_out4, int n_nodes) {
    int t = blockIdx.x * blockDim.x + threadIdx.x;
    if (t < n_out4) out4[t] = make_float4(0.f, 0.f, 0.f, 0.f);
    if (t < n_nodes) deg[t] = 0.f;
}

// ---------------------------------------------------------------------------
// Kernel: out = feat + (deg > 0 ? nsum / deg : 0), vectorized float4.
// One thread per 4 elements; 32 chunks per node row.
// ---------------------------------------------------------------------------
__global__ __launch_bounds__(256) void sage_finalize_kernel(
    const float* __restrict__ feat,
    const float* __restrict__ deg,
    float* __restrict__ out,
    int n_vec4) {
    int t = blockIdx.x * blockDim.x + threadIdx.x;
    if (t >= n_vec4) return;

    const int node = t >> 5;  // D_FEAT / 4 == 32 float4 chunks per node
    const float dg = deg[node];
    const float inv = (dg > 0.f) ? (1.0f / dg) : 0.0f;

    float4 s = *reinterpret_cast<float4*>(out + (size_t)t * 4);
    const float4 f = *reinterpret_cast<const float4*>(feat + (size_t)t * 4);

    s.x = fmaf(s.x, inv, f.x);
    s.y = fmaf(s.y, inv, f.y);
    s.z = fmaf(s.z, inv, f.z);
    s.w = fmaf(s.w, inv, f.w);

    *reinterpret_cast<float4*>(out + (size_t)t * 4) = s;
}

// ---------------------------------------------------------------------------
// Launcher
// ---------------------------------------------------------------------------
extern "C" void kernel_launch(void* const* d_in, const int* in_sizes, int n_in,
                              void* d_out, int out_size, void* d_ws, size_t ws_size,
                              hipStream_t stream) {
    const float* feat = (const float*)d_in[0];
    const int* src = (const int*)d_in[1];
    const int* dst = (const int*)d_in[2];

    const int n_feat = in_sizes[0];        // N * 128
    const int n_edges = in_sizes[1];       // E
    const int n_nodes = n_feat / D_FEAT;   // N

    float* out = (float*)d_out;            // doubles as nsum accumulator
    float* deg = (float*)d_ws;             // N floats of scratch

    const int n_out4 = n_feat / 4;

    // 1) zero accumulator + degrees
    {
        const int threads = 256;
        const int work = (n_out4 > n_nodes) ? n_out4 : n_nodes;
        const int blocks = (work + threads - 1) / threads;
        sage_init_kernel<<<blocks, threads, 0, stream>>>(
            (float4*)out, deg, n_out4, n_nodes);
    }

    // 2) edge scatter: one wave32 per edge (8 edges per 256-thread block)
    {
        const int threads = 256;
        const int waves_per_block = threads / 32;
        const int blocks = (n_edges + waves_per_block - 1) / waves_per_block;
        sage_edge_scatter_kernel<<<blocks, threads, 0, stream>>>(
            feat, src, dst, out, deg, n_edges);
    }

    // 3) finalize: residual + mean normalization
    {
        const int threads = 256;
        const int blocks = (n_out4 + threads - 1) / threads;
        sage_finalize_kernel<<<blocks, threads, 0, stream>>>(
            feat, deg, out, n_out4);
    }
}